// TopKTopPSampler_4045859193160
// MI455X (gfx1250) — compile-verified
//
#include <hip/hip_runtime.h>
#include <hip/hip_bf16.h>
#include <stdint.h>

// Top-k + top-p (nucleus) sampling, B=128 rows x V=128000 vocab.
// One workgroup per row; TDM (tensor_load_to_lds) double-buffered streaming.

#define BLOCK   256
#define TILE_F  4096          // floats per LDS tile (16 KB)
#define CAP     512           // candidate buffer (power of two for bitonic sort)
#define KMAX    64            // reference guarantees k <= 64

typedef __attribute__((ext_vector_type(4))) unsigned int u32x4;
typedef __attribute__((ext_vector_type(8))) int          i32x8;
typedef __attribute__((ext_vector_type(4))) int          i32x4;

#if defined(__AMDGCN__) && __has_builtin(__builtin_amdgcn_tensor_load_to_lds)
#define USE_TDM 1
#else
#define USE_TDM 0
#endif

#if __has_builtin(__builtin_amdgcn_s_wait_tensorcnt)
#define WAIT_TENSORCNT(n) __builtin_amdgcn_s_wait_tensorcnt((short)(n))
#else
#define WAIT_TENSORCNT(n) asm volatile("s_wait_tensorcnt %0" :: "i"(n) : "memory")
#endif

#if USE_TDM
// Build a 1-D TDM descriptor (D#) per CDNA5 ISA ch.8 and issue tensor_load_to_lds.
// group0: count=1 | lds_addr | global_addr[56:0] | type=2
// group1: data_size=4B, tensor_dim0=n, tensor_dim1=1, tile_dim0=n, tile_dim1=1
__device__ __forceinline__ void tdm_load_1d(unsigned lds_byte_off,
                                            const float* gsrc, unsigned nelem) {
  unsigned long long ga = (unsigned long long)(uintptr_t)gsrc;
  u32x4 g0;
  g0[0] = 1u;                                              // count=1, is_restore=0, gather=0
  g0[1] = lds_byte_off;                                    // lds_addr (bytes)
  g0[2] = (unsigned)(ga & 0xFFFFFFFFull);                  // global_addr[31:0]
  g0[3] = ((unsigned)(ga >> 32) & 0x01FFFFFFu) | (2u << 30); // addr[56:32] | type=2
  i32x8 g1;
  g1[0] = (int)(2u << 16);                                 // data_size=2 (4 bytes)
  g1[1] = (int)((nelem & 0xFFFFu) << 16);                  // tensor_dim0[15:0] @ [31:16]
  g1[2] = (int)(((nelem >> 16) & 0xFFFFu) | (1u << 16));   // tensor_dim0[31:16] | tensor_dim1=1
  g1[3] = (int)((nelem & 0xFFFFu) << 16);                  // tensor_dim1 hi=0 | tile_dim0=n
  g1[4] = 1;                                               // tile_dim1=1, tile_dim2=0
  g1[5] = (int)nelem;                                      // tensor_dim0_stride[31:0]
  g1[6] = 0;
  g1[7] = 0;
  i32x4 g2 = {0, 0, 0, 0};
  i32x4 g3 = {0, 0, 0, 0};
  i32x8 g4 = {0, 0, 0, 0, 0, 0, 0, 0};                     // clang-23 6-arg form extra group
  __builtin_amdgcn_tensor_load_to_lds(g0, g1, g2, g3, g4, 0);
}
#endif

// Stream one row through double-buffered LDS tiles, calling f(x, index).
template <typename F>
__device__ __forceinline__ void stream_row(const float* __restrict__ row, int V,
                                           float* t0, float* t1, F f) {
  const int ntiles = (V + TILE_F - 1) / TILE_F;
#if USE_TDM
  const unsigned off0 = (unsigned)(uintptr_t)t0;   // low 32 bits of flat addr == LDS offset
  const unsigned off1 = (unsigned)(uintptr_t)t1;
  if (threadIdx.x < 32u) {                          // wave 0 drives the DMA
    tdm_load_1d(off0, row, (unsigned)min(TILE_F, V));
  }
  for (int t = 0; t < ntiles; ++t) {
    const float* cur = (t & 1) ? t1 : t0;
    if (threadIdx.x < 32u) {
      if (t + 1 < ntiles) {
        tdm_load_1d((t & 1) ? off0 : off1, row + (t + 1) * TILE_F,
                    (unsigned)min(TILE_F, V - (t + 1) * TILE_F));
        WAIT_TENSORCNT(1);                          // current tile landed, next in flight
      } else {
        WAIT_TENSORCNT(0);
      }
    }
    __syncthreads();
    const int base = t * TILE_F;
    const int cnt  = min(TILE_F, V - base);
    for (int j = (int)threadIdx.x; j < cnt; j += BLOCK) f(cur[j], base + j);
    asm volatile("s_wait_dscnt 0x0" ::: "memory");  // all LDS reads retired before reuse
    __syncthreads();
  }
#else
  (void)t0; (void)t1;
  for (int t = 0; t < ntiles; ++t) {
    const int base = t * TILE_F;
    const int cnt  = min(TILE_F, V - base);
    for (int j = (int)threadIdx.x; j < cnt; j += BLOCK) f(row[base + j], base + j);
    __syncthreads();
  }
#endif
}

__global__ __launch_bounds__(BLOCK) void topk_topp_sample_kernel(
    const float* __restrict__ logits, const int* __restrict__ kArr,
    const float* __restrict__ pArr, float* __restrict__ out, int V) {
  __shared__ __align__(16) float tile0[TILE_F];
  __shared__ __align__(16) float tile1[TILE_F];
  __shared__ unsigned hist[256];
  __shared__ float redA[BLOCK];
  __shared__ float redB[BLOCK];
  __shared__ int   redI[BLOCK];
  __shared__ float candVal[CAP];
  __shared__ int   candIdx[CAP];
  __shared__ unsigned candCount;
  __shared__ int   sBstar;
  __shared__ float sM, sS;

  const int row = (int)blockIdx.x;
  const int tid = (int)threadIdx.x;
  const float* rowPtr = logits + (long long)row * V;

  // ---- Pass A (HBM): online softmax -> row max M, sum S ----
  float m = -INFINITY, s = 0.0f;
  stream_row(rowPtr, V, tile0, tile1, [&](float x, int) {
    if (x > m) { s = s * __expf(m - x) + 1.0f; m = x; }
    else       { s += __expf(x - m); }
  });
  redA[tid] = m; redB[tid] = s;
  __syncthreads();
  for (int off = BLOCK / 2; off > 0; off >>= 1) {
    if (tid < off) {
      float m1 = redA[tid],       s1 = redB[tid];
      float m2 = redA[tid + off], s2 = redB[tid + off];
      float Mx = fmaxf(m1, m2);
      float Sx = (s1 > 0.f ? s1 * __expf(m1 - Mx) : 0.f) +
                 (s2 > 0.f ? s2 * __expf(m2 - Mx) : 0.f);
      redA[tid] = Mx; redB[tid] = Sx;
    }
    __syncthreads();
  }
  if (tid == 0) { sM = redA[0]; sS = redB[0]; }
  __syncthreads();
  const float M = sM;
  const float S = sS;

  // ---- Pass B (L2): grow window until count(x >= M - delta) >= KMAX ----
  float delta = 0.75f;
  int total = 0;
  for (int it = 0; it < 9; ++it) {
    const float thr = M - delta;
    int c = 0;
    stream_row(rowPtr, V, tile0, tile1, [&](float x, int) { c += (x >= thr) ? 1 : 0; });
    redI[tid] = c;
    __syncthreads();
    for (int off = BLOCK / 2; off > 0; off >>= 1) {
      if (tid < off) redI[tid] += redI[tid + off];
      __syncthreads();
    }
    total = redI[0];
    __syncthreads();
    if (total >= KMAX) break;
    delta *= 2.5f;
  }
  if (total < KMAX) delta = 2.0e38f;     // degenerate spread: window = everything
  const float th  = M - delta;
  const float inv = 256.0f / delta;

  // ---- Pass C (L2): 256-bin histogram over window, pick 64th-largest bin ----
  hist[tid] = 0u;
  __syncthreads();
  stream_row(rowPtr, V, tile0, tile1, [&](float x, int) {
    if (x >= th) {
      int b = (int)((x - th) * inv);
      b = b < 0 ? 0 : (b > 255 ? 255 : b);
      atomicAdd(&hist[b], 1u);
    }
  });
  if (tid == 0) {
    unsigned acc = 0; int b = 255;
    for (; b > 0; --b) { acc += hist[b]; if (acc >= KMAX) break; }
    sBstar = b;
  }
  for (int e = tid; e < CAP; e += BLOCK) { candVal[e] = -INFINITY; candIdx[e] = 0; }
  if (tid == 0) candCount = 0u;
  __syncthreads();
  const int bstar = sBstar;

  // ---- Pass D (L2): collect candidates at/above threshold bin ----
  stream_row(rowPtr, V, tile0, tile1, [&](float x, int idx) {
    if (x >= th) {
      int b = (int)((x - th) * inv);
      b = b < 0 ? 0 : (b > 255 ? 255 : b);
      if (b >= bstar) {
        unsigned pos = atomicAdd(&candCount, 1u);
        if (pos < CAP) { candVal[pos] = x; candIdx[pos] = idx; }
      }
    }
  });
  __syncthreads();

  // ---- Bitonic sort descending (512 slots, padded with -inf) ----
  for (int kk2 = 2; kk2 <= CAP; kk2 <<= 1) {
    for (int j = kk2 >> 1; j > 0; j >>= 1) {
      for (int e = tid; e < CAP; e += BLOCK) {
        const int ixj = e ^ j;
        if (ixj > e) {
          float a = candVal[e], b = candVal[ixj];
          const bool up = ((e & kk2) == 0);
          const bool doSwap = up ? (a < b) : (a > b);
          if (doSwap) {
            candVal[e] = b; candVal[ixj] = a;
            int t2 = candIdx[e]; candIdx[e] = candIdx[ixj]; candIdx[ixj] = t2;
          }
        }
      }
      __syncthreads();
    }
  }

  // ---- Finalize: top-k AND top-p prefix, inverse-CDF categorical sample ----
  if (tid == 0) {
    int n = (int)candCount; if (n > CAP) n = CAP;
    int kk = kArr[row]; kk = kk < 1 ? 1 : (kk > KMAX ? KMAX : kk);
    const float pp = pArr[row];
    int r = 0; float csum = 0.0f;
    for (int i = 0; i < n; ++i) {
      if (i >= kk || csum >= pp) break;    // keep iff rank<k and csum_before<p
      csum += __expf(candVal[i] - M) / S;
      r = i + 1;
    }
    if (r < 1) r = 1;
    // deterministic uniform keyed on (seed=42, row) — splitmix64
    unsigned long long z = 42ull * 0x9E3779B97F4A7C15ull + (unsigned long long)(row + 1);
    z ^= z >> 30; z *= 0xBF58476D1CE4E5B9ull;
    z ^= z >> 27; z *= 0x94D049BB133111EBull;
    z ^= z >> 31;
    const float u = (float)(unsigned)(z >> 40) * (1.0f / 16777216.0f);
    const float target = u * csum;
    float acc = 0.0f; int tok = candIdx[0];
    for (int i = 0; i < r; ++i) {
      acc += __expf(candVal[i] - M) / S;
      tok = candIdx[i];
      if (acc > target) break;
    }
    out[row] = (float)tok;
  }
}

extern "C" void kernel_launch(void* const* d_in, const int* in_sizes, int n_in,
                              void* d_out, int out_size, void* d_ws, size_t ws_size,
                              hipStream_t stream) {
  (void)n_in; (void)d_ws; (void)ws_size; (void)out_size;
  const float* logits = (const float*)d_in[0];
  const int*   kArr   = (const int*)d_in[1];
  const float* pArr   = (const float*)d_in[2];
  float* out = (float*)d_out;
  const int B = in_sizes[1] > 0 ? in_sizes[1] : 1;
  const int V = in_sizes[0] / B;
  topk_topp_sample_kernel<<<dim3((unsigned)B), dim3(BLOCK), 0, stream>>>(
      logits, kArr, pArr, out, V);
}